// MaskEncoding_39307540693367
// MI455X (gfx1250) — compile-verified
//
#include <hip/hip_runtime.h>
#include <hip/hip_bf16.h>

// MaskEncoding: out[n,t,c] = (apply_mask[n] && rm[n] <= t < rm[n]+200) ? 0 : in[n,t,c]
// Shapes: waveforms [N=1024, T=2000, C=64] f32, apply_mask [N] i32, rm [N] i32.
// Pure streaming op: ~1 GiB moved, HBM-bound (~45 us at 23.3 TB/s). No FLOPs -> no WMMA.
// Strategy: b128 vectorized (native clang vector for nontemporal builtins),
// non-temporal cache hints (CDNA5 TH field) to avoid thrashing 192MB L2 with a
// 1 GiB stream, wave-uniform scalar loads of per-sample mask params, and no
// read at all on masked rows (store zeros directly).

typedef float v4f __attribute__((ext_vector_type(4)));

#ifndef MASK_T
#define MASK_T 2000
#endif
#ifndef MASK_LEN
#define MASK_LEN 200
#endif
#define VEC_PER_SAMPLE (MASK_T * 16)   // 16 x float4 per (n,t) row of 64 floats

__global__ __launch_bounds__(256)
void MaskEncoding_kernel(const v4f* __restrict__ in,
                         const int* __restrict__ apply_mask,
                         const int* __restrict__ rm,
                         v4f*       __restrict__ out) {
    const int n = blockIdx.y;                              // sample index (uniform per block)
    const int i = blockIdx.x * 256 + threadIdx.x;          // float4 index within sample, 0..31999
    const int t = i >> 4;                                  // time index, 0..1999

    // Wave-uniform: lowers to s_load_b32 (SMEM path).
    const int am = apply_mask[n];
    const int r  = rm[n];

    const long long idx = (long long)n * VEC_PER_SAMPLE + i;

    const bool masked = (am != 0) & ((unsigned)(t - r) < (unsigned)MASK_LEN);

    v4f v;
    if (masked) {
        v = (v4f){0.0f, 0.0f, 0.0f, 0.0f};                 // skip the read entirely
    } else {
        v = __builtin_nontemporal_load(&in[idx]);          // global_load_b128 th:NT
    }
    __builtin_nontemporal_store(v, &out[idx]);             // global_store_b128 th:NT
}

extern "C" void kernel_launch(void* const* d_in, const int* in_sizes, int n_in,
                              void* d_out, int out_size, void* d_ws, size_t ws_size,
                              hipStream_t stream) {
    const v4f* in         = (const v4f*)d_in[0];           // waveforms [1024,2000,64] f32
    const int* apply_mask = (const int*)d_in[1];           // [1024] i32
    const int* rm         = (const int*)d_in[2];           // [1024] i32
    v4f*       out        = (v4f*)d_out;                   // [1024,2000,64] f32

    const int N = in_sizes[1];                             // 1024 samples
    dim3 grid(VEC_PER_SAMPLE / 256, N, 1);                 // (125, 1024)
    dim3 block(256, 1, 1);                                 // 8 wave32s per block
    MaskEncoding_kernel<<<grid, block, 0, stream>>>(in, apply_mask, rm, out);
}